// SMEAR_43868795962047
// MI455X (gfx1250) — compile-verified
//
#include <hip/hip_runtime.h>
#include <math.h>

typedef __attribute__((ext_vector_type(2))) float v2f;
typedef __attribute__((ext_vector_type(8))) float v8f;

#define H_DIM 1024
#define FF_DIM 4096
#define E_DIM 8
#define B_DIM 4
#define S_DIM 1024
#define NTOK (B_DIM * S_DIM)   // 4096

// ---------------- workspace layout (bytes) ----------------
#define OFF_EW   ((size_t)0)                       // 8 floats
#define OFF_RLN  ((size_t)4096)                    // 4*1024 floats (16 KB)
#define OFF_MB1  ((size_t)32768)                   // 4096 floats (16 KB)
#define OFF_MB2  ((size_t)49152)                   // 1024 floats (4 KB)
#define OFF_MW1  ((size_t)65536)                   // 1024*4096 floats (16 MB)
#define OFF_MW2  (OFF_MW1 + (size_t)H_DIM * FF_DIM * 4)
#define OFF_H    (OFF_MW2 + (size_t)H_DIM * FF_DIM * 4)  // 4096*4096 floats (64 MB)

__device__ __forceinline__ float gelu_tanh(float x) {
    // jax.nn.gelu default (tanh approximation)
    float x3 = x * x * x;
    return 0.5f * x * (1.0f + tanhf(0.7978845608028654f * (x + 0.044715f * x3)));
}

// ---------------------------------------------------------------------------
// Kernel 1: router_in = mean_s(inputs)  then LayerNorm.  grid=4, block=256.
// ---------------------------------------------------------------------------
__global__ void routing_mean_ln(const float* __restrict__ x,
                                const float* __restrict__ g,
                                const float* __restrict__ bb,
                                float* __restrict__ rln) {
    const int b = blockIdx.x;
    const int tid = threadIdx.x;
    float acc[4] = {0.f, 0.f, 0.f, 0.f};
    const float* base = x + (size_t)b * S_DIM * H_DIM;
    for (int s = 0; s < S_DIM; ++s) {
        const float* row = base + (size_t)s * H_DIM;
#pragma unroll
        for (int j = 0; j < 4; ++j) acc[j] += row[tid + j * 256];
    }
    float val[4];
    float psum = 0.f, psq = 0.f;
#pragma unroll
    for (int j = 0; j < 4; ++j) {
        val[j] = acc[j] * (1.0f / S_DIM);
        psum += val[j];
        psq += val[j] * val[j];
    }
    __shared__ float r_s[256];
    __shared__ float r_q[256];
    r_s[tid] = psum;
    r_q[tid] = psq;
    __syncthreads();
    for (int off = 128; off > 0; off >>= 1) {
        if (tid < off) {
            r_s[tid] += r_s[tid + off];
            r_q[tid] += r_q[tid + off];
        }
        __syncthreads();
    }
    const float mu = r_s[0] * (1.0f / H_DIM);
    const float var = r_q[0] * (1.0f / H_DIM) - mu * mu;
    const float inv = rsqrtf(var + 1e-5f);
#pragma unroll
    for (int j = 0; j < 4; ++j) {
        int h = tid + j * 256;
        rln[(size_t)b * H_DIM + h] = (val[j] - mu) * inv * g[h] + bb[h];
    }
}

// ---------------------------------------------------------------------------
// Kernel 2: normalized router logits -> softmax -> batch-mean expert weights.
// 1 block, 256 threads = 8 waves (wave32); wave e handles expert e.
// ---------------------------------------------------------------------------
__global__ void router_kernel(const float* __restrict__ rln,
                              const float* __restrict__ rw,
                              const float* __restrict__ rb,
                              float* __restrict__ ew) {
    const int tid = threadIdx.x;
    const int lane = tid & 31;
    const int e = tid >> 5;
    __shared__ float logits[B_DIM][E_DIM];
    __shared__ float probs[B_DIM][E_DIM];

    float sq = 0.f, d0 = 0.f, d1 = 0.f, d2 = 0.f, d3 = 0.f;
    for (int h = lane; h < H_DIM; h += 32) {
        float w = rw[e * H_DIM + h];
        sq += w * w;
        d0 += w * rln[0 * H_DIM + h];
        d1 += w * rln[1 * H_DIM + h];
        d2 += w * rln[2 * H_DIM + h];
        d3 += w * rln[3 * H_DIM + h];
    }
#pragma unroll
    for (int off = 16; off > 0; off >>= 1) {
        sq += __shfl_down(sq, off, 32);
        d0 += __shfl_down(d0, off, 32);
        d1 += __shfl_down(d1, off, 32);
        d2 += __shfl_down(d2, off, 32);
        d3 += __shfl_down(d3, off, 32);
    }
    if (lane == 0) {
        float inv = 1.0f / fmaxf(sqrtf(sq), 1e-12f);
        logits[0][e] = d0 * inv + rb[e];
        logits[1][e] = d1 * inv + rb[e];
        logits[2][e] = d2 * inv + rb[e];
        logits[3][e] = d3 * inv + rb[e];
    }
    __syncthreads();
    if (tid < B_DIM) {
        float m = -1e30f;
#pragma unroll
        for (int i = 0; i < E_DIM; ++i) m = fmaxf(m, logits[tid][i]);
        float s = 0.f;
#pragma unroll
        for (int i = 0; i < E_DIM; ++i) {
            float p = __expf(logits[tid][i] - m);
            probs[tid][i] = p;
            s += p;
        }
        float invs = 1.0f / s;
#pragma unroll
        for (int i = 0; i < E_DIM; ++i) probs[tid][i] *= invs;
    }
    __syncthreads();
    if (tid < E_DIM) {
        ew[tid] = 0.25f * (probs[0][tid] + probs[1][tid] + probs[2][tid] + probs[3][tid]);
    }
}

// ---------------------------------------------------------------------------
// Kernel 3: streaming weighted merge of expert params (bandwidth phase).
// float4 grid-stride over mw1 | mw2 | mb1 | mb2.  ~300 MB @ 23.3 TB/s.
// ---------------------------------------------------------------------------
__global__ void merge_kernel(const float* __restrict__ w1, const float* __restrict__ b1,
                             const float* __restrict__ w2, const float* __restrict__ b2,
                             const float* __restrict__ ew,
                             float* __restrict__ mw1, float* __restrict__ mb1,
                             float* __restrict__ mw2, float* __restrict__ mb2) {
    const size_t HF4 = (size_t)H_DIM * FF_DIM / 4;        // float4 count per matrix
    const size_t NB1 = FF_DIM / 4;
    const size_t NB2 = H_DIM / 4;
    const size_t total = 2 * HF4 + NB1 + NB2;
    float w[E_DIM];
#pragma unroll
    for (int i = 0; i < E_DIM; ++i) w[i] = ew[i];

    size_t stride = (size_t)gridDim.x * blockDim.x;
    for (size_t i = (size_t)blockIdx.x * blockDim.x + threadIdx.x; i < total; i += stride) {
        float ax = 0.f, ay = 0.f, az = 0.f, aw = 0.f;
        if (i < HF4) {
#pragma unroll
            for (int ee = 0; ee < E_DIM; ++ee) {
                const float4 v = ((const float4*)w1)[(size_t)ee * HF4 + i];
                ax += w[ee] * v.x; ay += w[ee] * v.y; az += w[ee] * v.z; aw += w[ee] * v.w;
            }
            ((float4*)mw1)[i] = make_float4(ax, ay, az, aw);
        } else if (i < 2 * HF4) {
            size_t j = i - HF4;
#pragma unroll
            for (int ee = 0; ee < E_DIM; ++ee) {
                const float4 v = ((const float4*)w2)[(size_t)ee * HF4 + j];
                ax += w[ee] * v.x; ay += w[ee] * v.y; az += w[ee] * v.z; aw += w[ee] * v.w;
            }
            ((float4*)mw2)[j] = make_float4(ax, ay, az, aw);
        } else if (i < 2 * HF4 + NB1) {
            size_t j = i - 2 * HF4;
#pragma unroll
            for (int ee = 0; ee < E_DIM; ++ee) {
                const float4 v = ((const float4*)b1)[(size_t)ee * NB1 + j];
                ax += w[ee] * v.x; ay += w[ee] * v.y; az += w[ee] * v.z; aw += w[ee] * v.w;
            }
            ((float4*)mb1)[j] = make_float4(ax, ay, az, aw);
        } else {
            size_t j = i - 2 * HF4 - NB1;
#pragma unroll
            for (int ee = 0; ee < E_DIM; ++ee) {
                const float4 v = ((const float4*)b2)[(size_t)ee * NB2 + j];
                ax += w[ee] * v.x; ay += w[ee] * v.y; az += w[ee] * v.z; aw += w[ee] * v.w;
            }
            ((float4*)mb2)[j] = make_float4(ax, ay, az, aw);
        }
    }
}

// ---------------------------------------------------------------------------
// Kernels 4/5: fp32 WMMA GEMM, C = [gelu](A[M,K] @ B[K,N] + bias[N]).
//
// Block tile 128x64 (BK=16), 256 threads = 8 waves in a 4x2 grid; each wave
// owns a 32x32 tile = 2x2 WMMA accumulators (v8f each).  Per K-step: 4
// fragment loads (a0,a1,b0,b1) feed 4 V_WMMA_F32_16X16X4_F32 (each fragment
// reused twice).  Global->LDS staging is software-pipelined: next tile's
// global_load_b128 issue before the WMMA burst of the current tile.
//
// 32-bit A (16x4) layout: lanes 0-15: V0=K0,V1=K1; lanes 16-31: V0=K2,V1=K3.
// B (4x16) symmetric; B staged transposed in LDS so fragments are contiguous
// float2 LDS loads.  C/D layout: lane L, vgpr v -> row (L/16)*8+v, col L%16.
// ---------------------------------------------------------------------------
template <bool GELU>
__global__ void gemm_wmma(const float* __restrict__ A, const float* __restrict__ B,
                          const float* __restrict__ bias, float* __restrict__ C,
                          int M, int N, int K) {
    __shared__ float As[128 * 16];  // [m][k], row-major, 8 KB
    __shared__ float Bs[64 * 16];   // [n][k] (transposed tile), 4 KB

    const int tid = threadIdx.x;
    const int lane = tid & 31;
    const int wave = tid >> 5;
    const int wm = wave & 3;        // 0..3 -> 32-row M strip
    const int wn = wave >> 2;       // 0..1 -> 32-col N strip
    const int half = lane >> 4;     // K-pair selector
    const int r = lane & 15;

    const int row0 = blockIdx.y * 128;
    const int n0 = blockIdx.x * 64;

    v8f c00 = {}, c01 = {}, c10 = {}, c11 = {};

    // per-thread staging coordinates:
    // A tile: 128x16 = 2048 floats -> two float4 per thread
    const int a_m0 = (tid * 4) >> 4;         // 0..63
    const int a_m1 = a_m0 + 64;              // 64..127
    const int a_k = (tid * 4) & 15;          // 0,4,8,12
    // B tile: 16x64 = 1024 floats -> one float4 per thread
    const int b_k = (tid * 4) >> 6;          // 0..15
    const int b_n = (tid * 4) & 63;          // 0..60 step 4

    float4 ra0, ra1, rbv;
    {
        ra0 = *(const float4*)&A[(size_t)(row0 + a_m0) * K + a_k];
        ra1 = *(const float4*)&A[(size_t)(row0 + a_m1) * K + a_k];
        rbv = *(const float4*)&B[(size_t)b_k * N + n0 + b_n];
    }

    for (int k0 = 0; k0 < K; k0 += 16) {
        // stage current tile into LDS
        *(float4*)&As[a_m0 * 16 + a_k] = ra0;
        *(float4*)&As[a_m1 * 16 + a_k] = ra1;
        Bs[(b_n + 0) * 16 + b_k] = rbv.x;
        Bs[(b_n + 1) * 16 + b_k] = rbv.y;
        Bs[(b_n + 2) * 16 + b_k] = rbv.z;
        Bs[(b_n + 3) * 16 + b_k] = rbv.w;
        __syncthreads();

        // prefetch next tile (overlaps the WMMA burst below)
        if (k0 + 16 < K) {
            ra0 = *(const float4*)&A[(size_t)(row0 + a_m0) * K + k0 + 16 + a_k];
            ra1 = *(const float4*)&A[(size_t)(row0 + a_m1) * K + k0 + 16 + a_k];
            rbv = *(const float4*)&B[(size_t)(k0 + 16 + b_k) * N + n0 + b_n];
        }

#pragma unroll
        for (int kk = 0; kk < 16; kk += 4) {
            v2f a0 = *(const v2f*)&As[(wm * 32 + r) * 16 + kk + half * 2];
            v2f a1 = *(const v2f*)&As[(wm * 32 + 16 + r) * 16 + kk + half * 2];
            v2f b0 = *(const v2f*)&Bs[(wn * 32 + r) * 16 + kk + half * 2];
            v2f b1 = *(const v2f*)&Bs[(wn * 32 + 16 + r) * 16 + kk + half * 2];
            c00 = __builtin_amdgcn_wmma_f32_16x16x4_f32(false, a0, false, b0,
                                                        (short)0, c00, false, false);
            c01 = __builtin_amdgcn_wmma_f32_16x16x4_f32(false, a0, false, b1,
                                                        (short)0, c01, false, false);
            c10 = __builtin_amdgcn_wmma_f32_16x16x4_f32(false, a1, false, b0,
                                                        (short)0, c10, false, false);
            c11 = __builtin_amdgcn_wmma_f32_16x16x4_f32(false, a1, false, b1,
                                                        (short)0, c11, false, false);
        }
        __syncthreads();
    }

    const int gc0 = n0 + wn * 32 + r;
    const int gc1 = gc0 + 16;
    const float bi0 = bias[gc0];
    const float bi1 = bias[gc1];
#pragma unroll
    for (int v = 0; v < 8; ++v) {
        int gr0 = row0 + wm * 32 + half * 8 + v;   // c00/c01 rows
        int gr1 = gr0 + 16;                        // c10/c11 rows
        float x00 = c00[v] + bi0;
        float x01 = c01[v] + bi1;
        float x10 = c10[v] + bi0;
        float x11 = c11[v] + bi1;
        if (GELU) {
            x00 = gelu_tanh(x00); x01 = gelu_tanh(x01);
            x10 = gelu_tanh(x10); x11 = gelu_tanh(x11);
        }
        C[(size_t)gr0 * N + gc0] = x00;
        C[(size_t)gr0 * N + gc1] = x01;
        C[(size_t)gr1 * N + gc0] = x10;
        C[(size_t)gr1 * N + gc1] = x11;
    }
}

// ---------------------------------------------------------------------------
extern "C" void kernel_launch(void* const* d_in, const int* in_sizes, int n_in,
                              void* d_out, int out_size, void* d_ws, size_t ws_size,
                              hipStream_t stream) {
    const float* inputs   = (const float*)d_in[0];  // [4,1024,1024]
    const float* ln_g     = (const float*)d_in[1];  // [1024]
    const float* ln_b     = (const float*)d_in[2];  // [1024]
    const float* router_w = (const float*)d_in[3];  // [8,1024]
    const float* router_b = (const float*)d_in[4];  // [8]
    const float* w1       = (const float*)d_in[5];  // [8,1024,4096]
    const float* b1       = (const float*)d_in[6];  // [8,4096]
    const float* w2       = (const float*)d_in[7];  // [8,4096,1024]
    const float* b2       = (const float*)d_in[8];  // [8,1024]
    float* out = (float*)d_out;                     // [4,1024,1024]

    char* ws = (char*)d_ws;
    float* ew  = (float*)(ws + OFF_EW);
    float* rln = (float*)(ws + OFF_RLN);
    float* mb1 = (float*)(ws + OFF_MB1);
    float* mb2 = (float*)(ws + OFF_MB2);
    float* mw1 = (float*)(ws + OFF_MW1);
    float* mw2 = (float*)(ws + OFF_MW2);
    float* hbuf = (float*)(ws + OFF_H);

    // 1) routing stats
    routing_mean_ln<<<dim3(B_DIM), dim3(256), 0, stream>>>(inputs, ln_g, ln_b, rln);
    // 2) expert weights
    router_kernel<<<dim3(1), dim3(256), 0, stream>>>(rln, router_w, router_b, ew);
    // 3) weighted merge of all expert parameters (bandwidth phase)
    merge_kernel<<<dim3(2048), dim3(256), 0, stream>>>(w1, b1, w2, b2, ew,
                                                       mw1, mb1, mw2, mb2);
    // 4) h = gelu(X @ mw1 + mb1)   [4096 x 4096]
    gemm_wmma<true><<<dim3(FF_DIM / 64, NTOK / 128), dim3(256), 0, stream>>>(
        inputs, mw1, mb1, hbuf, NTOK, FF_DIM, H_DIM);
    // 5) out = h @ mw2 + mb2       [4096 x 1024]
    gemm_wmma<false><<<dim3(H_DIM / 64, NTOK / 128), dim3(256), 0, stream>>>(
        hbuf, mw2, mb2, out, NTOK, H_DIM, FF_DIM);
}